// CoDAStyleAttention_8589934592317
// MI455X (gfx1250) — compile-verified
//
#include <hip/hip_runtime.h>

typedef float v2f __attribute__((ext_vector_type(2)));
typedef float v8f __attribute__((ext_vector_type(8)));

#define HD    64
#define NH    16
#define BN    4
#define HW    4096            // 64*64
#define NROWS 16384           // B*nH*hd = 16*16*64

// ---------------------------------------------------------------------------
// Kernel 1: mean-pool over H*W.  One 256-thread block per (b,nH,hd) row.
// Each thread sums 16 floats via 4 float4 loads -> wave reduce -> LDS reduce.
// ---------------------------------------------------------------------------
__global__ void pool_kernel(const float* __restrict__ x, float* __restrict__ pooled) {
    __shared__ float red[8];
    const int row = blockIdx.x;
    const int t   = threadIdx.x;
    const float4* p = (const float4*)(x + (size_t)row * HW);
    float s = 0.f;
#pragma unroll
    for (int i = 0; i < 4; ++i) {
        float4 v = p[t + 256 * i];
        s += v.x + v.y + v.z + v.w;
    }
#pragma unroll
    for (int m = 16; m >= 1; m >>= 1) s += __shfl_xor(s, m, 32);
    if ((t & 31) == 0) red[t >> 5] = s;
    __syncthreads();
    if (t == 0) {
        float tot = 0.f;
#pragma unroll
        for (int w = 0; w < 8; ++w) tot += red[w];
        pooled[row] = tot * (1.0f / (float)HW);
    }
}

// ---------------------------------------------------------------------------
// Kernel 2: full bottleneck-attention pipeline on the pooled tensor.
// One wave32 per batch b (16 blocks x 32 threads).  compress & expand GEMMs
// run on V_WMMA_F32_16X16X4_F32; attention (heads=2, dh=2) is one lane per
// (head, query-row); LayerNorm via wave shuffles.  Writes delta[b][16][64].
// ---------------------------------------------------------------------------
__global__ void attn_kernel(const float* __restrict__ pooled,
                            const float* __restrict__ cw,  const float* __restrict__ cb,
                            const float* __restrict__ ipw, const float* __restrict__ ipb,
                            const float* __restrict__ opw, const float* __restrict__ opb,
                            const float* __restrict__ ew,  const float* __restrict__ eb,
                            const float* __restrict__ lnw, const float* __restrict__ lnb,
                            const float* __restrict__ gate,
                            float* __restrict__ delta) {
    __shared__ float xp [16][64];   // pooled residual tile
    __shared__ float xc [16][4];    // after compress
    __shared__ float qkv[16][12];   // q|k|v
    __shared__ float ao [16][4];    // attention output
    __shared__ float xa [16][4];    // after out_proj
    __shared__ float xe [16][64];   // after expand

    const int b    = blockIdx.x;
    const int lane = threadIdx.x;       // 0..31, full wave
    const int half = lane >> 4;         // 0: lanes 0-15, 1: lanes 16-31
    const int l15  = lane & 15;

    // ---- load x_pooled[b] (16x64) into LDS: 256 float4s, 8 per lane
    {
        const float4* src = (const float4*)(pooled + (size_t)b * (NH * HD));
        float4* dst = (float4*)&xp[0][0];
#pragma unroll
        for (int i = 0; i < 8; ++i) dst[lane + 32 * i] = src[lane + 32 * i];
    }
    __syncthreads();

    // ---- compress: xc = xp(16x64) @ cw^T(64x4) + cb   [WMMA f32 16x16x4, 16 K-steps]
    // A frag: lanes 0-15 hold K=k0,k0+1 of row M=l15; lanes 16-31 hold K=k0+2,k0+3.
    // B frag (4x16): row K striped across lanes (same half split); cols >= BN zeroed.
    {
        v8f c = {};
        const int nn = (l15 < BN) ? l15 : 0;         // keep loads in-bounds
        const bool valid = (l15 < BN);
#pragma unroll
        for (int kk = 0; kk < 16; ++kk) {
            const int k0 = kk * 4 + half * 2;
            v2f a, bb;
            a.x = xp[l15][k0 + 0];
            a.y = xp[l15][k0 + 1];
            const float b0 = cw[nn * HD + k0 + 0];   // cw^T[K][N] = cw[N][K]
            const float b1 = cw[nn * HD + k0 + 1];
            bb.x = valid ? b0 : 0.f;
            bb.y = valid ? b1 : 0.f;
            c = __builtin_amdgcn_wmma_f32_16x16x4_f32(false, a, false, bb,
                                                      (short)0, c, false, false);
        }
        // C/D layout: lane n(0-15) VGPR r -> (M=r, N=n); lanes 16-31 -> (M=r+8, N=n-16)
        if (l15 < BN) {
#pragma unroll
            for (int r = 0; r < 8; ++r)
                xc[r + 8 * half][l15] = c[r] + cb[l15];
        }
    }
    __syncthreads();

    // ---- qkv = xc(16x4) @ ipw^T(4x12) + ipb : 192 outputs, 6 per lane
    {
#pragma unroll
        for (int i = 0; i < 6; ++i) {
            const int idx = lane * 6 + i;
            const int L = idx / 12, j = idx % 12;
            float s = ipb[j];
#pragma unroll
            for (int e = 0; e < BN; ++e) s += xc[L][e] * ipw[j * BN + e];
            qkv[L][j] = s;
        }
    }
    __syncthreads();

    // ---- attention: lane = (head h = half, query row qi = l15); dh = 2
    {
        const int h = half, qi = l15;
        const float scale = 0.70710678118654752f;     // 1/sqrt(dh=2)
        const float q0 = qkv[qi][2 * h + 0];
        const float q1 = qkv[qi][2 * h + 1];
        float sc[16];
        float mx = -3.0e38f;
#pragma unroll
        for (int j = 0; j < 16; ++j) {
            const float k0 = qkv[j][4 + 2 * h + 0];
            const float k1 = qkv[j][4 + 2 * h + 1];
            sc[j] = (q0 * k0 + q1 * k1) * scale;
            mx = fmaxf(mx, sc[j]);
        }
        float den = 0.f;
#pragma unroll
        for (int j = 0; j < 16; ++j) { sc[j] = __expf(sc[j] - mx); den += sc[j]; }
        const float inv = 1.0f / den;
        float o0 = 0.f, o1 = 0.f;
#pragma unroll
        for (int j = 0; j < 16; ++j) {
            const float p = sc[j] * inv;
            o0 += p * qkv[j][8 + 2 * h + 0];
            o1 += p * qkv[j][8 + 2 * h + 1];
        }
        ao[qi][2 * h + 0] = o0;
        ao[qi][2 * h + 1] = o1;
    }
    __syncthreads();

    // ---- out_proj: xa = ao(16x4) @ opw^T(4x4) + opb : 64 outputs, 2 per lane
    {
#pragma unroll
        for (int i = 0; i < 2; ++i) {
            const int idx = lane * 2 + i;
            const int L = idx >> 2, n = idx & 3;
            float s = opb[n];
#pragma unroll
            for (int e = 0; e < BN; ++e) s += ao[L][e] * opw[n * BN + e];
            xa[L][n] = s;
        }
    }
    __syncthreads();

    // ---- expand: xe = xa(16x4) @ ew^T(4x64) + eb   [4 x WMMA f32 16x16x4, one K-step]
    {
        v2f a;
        a.x = xa[l15][2 * half + 0];
        a.y = xa[l15][2 * half + 1];
#pragma unroll
        for (int nt = 0; nt < 4; ++nt) {
            const int n = nt * 16 + l15;
            v2f bb;
            bb.x = ew[n * BN + 2 * half + 0];         // ew^T[K][n] = ew[n][K]
            bb.y = ew[n * BN + 2 * half + 1];
            v8f c = {};
            c = __builtin_amdgcn_wmma_f32_16x16x4_f32(false, a, false, bb,
                                                      (short)0, c, false, false);
#pragma unroll
            for (int r = 0; r < 8; ++r)
                xe[r + 8 * half][n] = c[r] + eb[n];
        }
    }
    __syncthreads();

    // ---- LayerNorm(residual + gate*expand) - residual -> delta
    {
        const float g = gate[0];
#pragma unroll 1
        for (int r = 0; r < 16; ++r) {
            const float x0 = xp[r][lane],      x1 = xp[r][lane + 32];
            const float y0 = x0 + g * xe[r][lane];
            const float y1 = x1 + g * xe[r][lane + 32];
            float s = y0 + y1;
#pragma unroll
            for (int m = 16; m >= 1; m >>= 1) s += __shfl_xor(s, m, 32);
            const float mu = s * (1.0f / 64.0f);
            const float d0 = y0 - mu, d1 = y1 - mu;
            float ss = d0 * d0 + d1 * d1;
#pragma unroll
            for (int m = 16; m >= 1; m >>= 1) ss += __shfl_xor(ss, m, 32);
            const float rstd = rsqrtf(ss * (1.0f / 64.0f) + 1e-5f);
            const float out0 = d0 * rstd * lnw[lane]      + lnb[lane];
            const float out1 = d1 * rstd * lnw[lane + 32] + lnb[lane + 32];
            float* drow = delta + (size_t)b * (NH * HD) + r * HD;
            drow[lane]      = out0 - x0;
            drow[lane + 32] = out1 - x1;
        }
    }
}

// ---------------------------------------------------------------------------
// Kernel 3: out = x + delta broadcast over H*W.  float4 streaming.
// ---------------------------------------------------------------------------
__global__ void bcast_add_kernel(const float* __restrict__ x,
                                 const float* __restrict__ delta,
                                 float* __restrict__ out) {
    const size_t i = (size_t)blockIdx.x * blockDim.x + threadIdx.x;  // float4 index
    const float4* x4 = (const float4*)x;
    float4* o4 = (float4*)out;
    float4 v = x4[i];
    const int row = (int)((i * 4) >> 12);   // element / 4096 -> (b,nH,hd) row
    const float d = delta[row];
    v.x += d; v.y += d; v.z += d; v.w += d;
    o4[i] = v;
}

// ---------------------------------------------------------------------------
extern "C" void kernel_launch(void* const* d_in, const int* in_sizes, int n_in,
                              void* d_out, int out_size, void* d_ws, size_t ws_size,
                              hipStream_t stream) {
    const float* x    = (const float*)d_in[0];
    const float* cw   = (const float*)d_in[1];
    const float* cb   = (const float*)d_in[2];
    const float* ipw  = (const float*)d_in[3];
    const float* ipb  = (const float*)d_in[4];
    const float* opw  = (const float*)d_in[5];
    const float* opb  = (const float*)d_in[6];
    const float* ew   = (const float*)d_in[7];
    const float* eb   = (const float*)d_in[8];
    const float* lnw  = (const float*)d_in[9];
    const float* lnb  = (const float*)d_in[10];
    const float* gate = (const float*)d_in[11];

    float* pooled = (float*)d_ws;           // NROWS floats
    float* delta  = pooled + NROWS;         // NROWS floats

    // Pass 1: pool 256 MB -> 64 KB
    pool_kernel<<<NROWS, 256, 0, stream>>>(x, pooled);

    // Pass 2: tiny attention pipeline, one wave32 per batch
    attn_kernel<<<16, 32, 0, stream>>>(pooled, cw, cb, ipw, ipb, opw, opb,
                                       ew, eb, lnw, lnb, gate, delta);

    // Pass 3: broadcast add, 16M float4s
    const size_t n4 = (size_t)NROWS * HW / 4;           // 16,777,216
    bcast_add_kernel<<<(unsigned)(n4 / 256), 256, 0, stream>>>(x, delta, (float*)d_out);
}